// DisplacementVectorsASU_67559835566336
// MI455X (gfx1250) — compile-verified
//
#include <hip/hip_runtime.h>
#include <cstdint>

#define TILE 128   // threads per block = edges per tile (4 wave32 waves)
#define SLOT 20    // floats per LDS slot: 12 data + 8 pad -> conflict-free ds_load_b128

__global__ __launch_bounds__(TILE, 1)
void disp_kernel(const float* __restrict__ frac,   // (N,3)
                 const int*   __restrict__ eidx,   // (2,M)
                 const float* __restrict__ symm,   // (M,4,4)
                 const float* __restrict__ cell,   // (M,3)
                 float*       __restrict__ out,    // (M,3)
                 int M, int ntiles)
{
    __shared__ __align__(16) float smem[2][TILE * SLOT];
    const int tid = threadIdx.x;

    // LDS byte offsets of this lane's two staging slots
    unsigned ldsSlot[2];
    ldsSlot[0] = (unsigned)(size_t)&smem[0][tid * SLOT];
    ldsSlot[1] = (unsigned)(size_t)&smem[1][tid * SLOT];

    const unsigned long long sbase = (unsigned long long)(uintptr_t)symm;

    long t = (long)blockIdx.x;
    if (t >= ntiles) return;

    // Prologue: async-stage rows 0..2 (48B) of tile t's symmops into LDS buf 0.
    // NT hint: symmops is a 256MB stream-once — keep it out of the 192MB L2 so
    // the frac_coords gather table (1.2MB, ~80x reuse) stays resident.
    {
        long e = t * TILE + tid;
        unsigned voff = (e < (long)M) ? (unsigned)(e * 64) : 0u;
        asm volatile(
            "global_load_async_to_lds_b128 %0, %1, %2 th:TH_LOAD_NT\n\t"
            "global_load_async_to_lds_b128 %0, %1, %2 offset:16 th:TH_LOAD_NT\n\t"
            "global_load_async_to_lds_b128 %0, %1, %2 offset:32 th:TH_LOAD_NT"
            :: "v"(ldsSlot[0]), "v"(voff), "s"(sbase)
            : "memory");
    }

    int buf = 0;
    for (; t < ntiles; t += gridDim.x) {
        const long tn   = t + (long)gridDim.x;
        const bool more = (tn < ntiles);
        if (more) {
            // Prefetch next tile's symmop rows into the other buffer.
            long e = tn * TILE + tid;
            unsigned voff = (e < (long)M) ? (unsigned)(e * 64) : 0u;
            unsigned dst  = ldsSlot[buf ^ 1];
            asm volatile(
                "global_load_async_to_lds_b128 %0, %1, %2 th:TH_LOAD_NT\n\t"
                "global_load_async_to_lds_b128 %0, %1, %2 offset:16 th:TH_LOAD_NT\n\t"
                "global_load_async_to_lds_b128 %0, %1, %2 offset:32 th:TH_LOAD_NT"
                :: "v"(dst), "v"(voff), "s"(sbase)
                : "memory");
            // 3 newer loads outstanding; async loads complete in order,
            // so asynccnt<=3 means this tile's 3 loads are done.
            asm volatile("s_wait_asynccnt 0x3" ::: "memory");
        } else {
            asm volatile("s_wait_asynccnt 0x0" ::: "memory");
        }

        const long e = t * TILE + tid;
        if (e < (long)M) {
            const float4* slot = (const float4*)&smem[buf][tid * SLOT];
            float4 r0 = slot[0];   // symmop row 0
            float4 r1 = slot[1];   // symmop row 1
            float4 r2 = slot[2];   // symmop row 2

            int i0 = __builtin_nontemporal_load(&eidx[e]);
            int i1 = __builtin_nontemporal_load(&eidx[(long)M + e]);
            const float* p0 = frac + 3 * (long)i0;   // in_frac  (keep L2-resident: RT)
            const float* p1 = frac + 3 * (long)i1;   // out_frac (keep L2-resident: RT)
            float inx = p0[0], iny = p0[1], inz = p0[2];
            float hx  = p1[0], hy  = p1[1], hz  = p1[2];

            // out_t[k] = dot(out_h, S[k][:]) with out_h = (hx,hy,hz,1)
            float o0 = fmaf(r0.x, hx, fmaf(r0.y, hy, fmaf(r0.z, hz, r0.w)));
            float o1 = fmaf(r1.x, hx, fmaf(r1.y, hy, fmaf(r1.z, hz, r1.w)));
            float o2 = fmaf(r2.x, hx, fmaf(r2.y, hy, fmaf(r2.z, hz, r2.w)));

            const long eb = 3 * e;
            float c0 = __builtin_nontemporal_load(&cell[eb]);
            float c1 = __builtin_nontemporal_load(&cell[eb + 1]);
            float c2 = __builtin_nontemporal_load(&cell[eb + 2]);
            o0 = o0 - floorf(o0) + c0;
            o1 = o1 - floorf(o1) + c1;
            o2 = o2 - floorf(o2) + c2;

            __builtin_nontemporal_store(inx - o0, &out[eb]);
            __builtin_nontemporal_store(iny - o1, &out[eb + 1]);
            __builtin_nontemporal_store(inz - o2, &out[eb + 2]);
        }
        buf ^= 1;
    }
}

extern "C" void kernel_launch(void* const* d_in, const int* in_sizes, int n_in,
                              void* d_out, int out_size, void* d_ws, size_t ws_size,
                              hipStream_t stream) {
    const float* frac = (const float*)d_in[0];   // (N,3) f32
    const int*   eidx = (const int*)  d_in[1];   // (2,M) i32
    const float* symm = (const float*)d_in[2];   // (M,4,4) f32
    const float* cell = (const float*)d_in[3];   // (M,3) f32
    float* out = (float*)d_out;                  // (M,3) f32

    const int M = in_sizes[1] / 2;
    if (M <= 0) return;
    const int ntiles = (M + TILE - 1) / TILE;
    int grid = ntiles < 4096 ? ntiles : 4096;

    disp_kernel<<<grid, TILE, 0, stream>>>(frac, eidx, symm, cell, out, M, ntiles);
}